// Defaultloss_49873160241482
// MI455X (gfx1250) — compile-verified
//
#include <hip/hip_runtime.h>

#define NB 16
#define NA 120000
#define NC 20
#define NG 64

typedef __attribute__((ext_vector_type(2))) float v2f;
typedef __attribute__((ext_vector_type(8))) float v8f;

__global__ void ws_init(float* ws) {
  if (threadIdx.x < 32) ws[threadIdx.x] = 0.0f;
}

__device__ __forceinline__ float smoothl1(float d) {
  const float beta = 1.0f / 9.0f;
  return (d <= beta) ? (4.5f * d * d) : (d - 0.5f * beta);
}

__global__ __launch_bounds__(256) void loss_kernel(
    const float* __restrict__ dt, const float* __restrict__ gt,
    const float* __restrict__ anchors, const int* __restrict__ assign,
    float* __restrict__ ws) {
  __shared__ float sgt[NG * 5];
  __shared__ float s_loss[8];
  __shared__ float s_cnt[8];

  const int b = blockIdx.y;
  const int tid = threadIdx.x;
  const int lane = tid & 31;
  const int wave = tid >> 5;

  for (int i = tid; i < NG * 5; i += 256) sgt[i] = gt[b * NG * 5 + i];
  __syncthreads();

  const float* __restrict__ dtb = dt + (size_t)b * (5 + NC) * NA;
  const int* __restrict__ asb = assign + (size_t)b * NA;

  v8f acc = {0.f, 0.f, 0.f, 0.f, 0.f, 0.f, 0.f, 0.f};
  const v2f ones = {1.0f, 1.0f};
  float cnt = 0.0f;

  const int gwave = blockIdx.x * 8 + wave;          // wave-uniform
  const int stride = gridDim.x * 8 * 128;

  for (int base = gwave * 128; base < NA; base += stride) {  // uniform bound
    v2f av = {0.0f, 0.0f};
    const int i0 = base + 4 * lane;
    if (i0 < NA) {
      // CDNA5 global_prefetch of the objectness stream's cacheline
      __builtin_prefetch(dtb + (size_t)4 * NA + i0, 0, 0);

      const int4 a4 = *(const int4*)(asb + i0);
      const int asg[4] = {a4.x, a4.y, a4.z, a4.w};

      float mcls[4], mbox[4], tg0[4], tg1[4], tg2[4], tg3[4];
      int cls[4];
#pragma unroll
      for (int j = 0; j < 4; ++j) {
        const int a = asg[j];
        mcls[j] = (a >= 0) ? 1.0f : 0.0f;
        mbox[j] = (a >= 1) ? 1.0f : 0.0f;
        cnt += mbox[j];
        int gi = a - 1;
        gi = gi < 0 ? 0 : (gi > NG - 1 ? NG - 1 : gi);
        const float g0 = sgt[gi * 5 + 0];
        const float g1 = sgt[gi * 5 + 1];
        const float gw = sgt[gi * 5 + 2];
        const float gh = sgt[gi * 5 + 3];
        cls[j] = (int)sgt[gi * 5 + 4] - 1;
        const float4 an = *(const float4*)(anchors + 4 * (i0 + j));
        const float aw = an.z - an.x;
        const float ah = an.w - an.y;
        const float ax = an.x + 0.5f * aw;
        const float ay = an.y + 0.5f * ah;
        tg0[j] = (g0 + 0.5f * gw - ax) / aw;
        tg1[j] = (g1 + 0.5f * gh - ay) / ah;
        tg2[j] = __logf(gw / aw);
        tg3[j] = __logf(gh / ah);
      }

      float v[4] = {0.f, 0.f, 0.f, 0.f};
#pragma unroll
      for (int ch = 0; ch <= NC; ++ch) {   // obj + C class channels
        const float4 p4 = *(const float4*)(dtb + (size_t)(4 + ch) * NA + i0);
        const float pv[4] = {p4.x, p4.y, p4.z, p4.w};
#pragma unroll
        for (int j = 0; j < 4; ++j) {
          const float p = fminf(fmaxf(pv[j], 1e-4f), 1.0f - 1e-4f);
          const bool pos = (mbox[j] != 0.0f) && ((ch == 0) || ((ch - 1) == cls[j]));
          const float pp = pos ? p : (1.0f - p);      // one log per channel
          const float om = 1.0f - pp;
          v[j] += (pos ? 0.25f : 0.75f) * om * om * (-__logf(pp));
        }
      }

      const float4 r0 = *(const float4*)(dtb + (size_t)0 * NA + i0);
      const float4 r1 = *(const float4*)(dtb + (size_t)1 * NA + i0);
      const float4 r2 = *(const float4*)(dtb + (size_t)2 * NA + i0);
      const float4 r3 = *(const float4*)(dtb + (size_t)3 * NA + i0);
      const float rr0[4] = {r0.x, r0.y, r0.z, r0.w};
      const float rr1[4] = {r1.x, r1.y, r1.z, r1.w};
      const float rr2[4] = {r2.x, r2.y, r2.z, r2.w};
      const float rr3[4] = {r3.x, r3.y, r3.z, r3.w};
#pragma unroll
      for (int j = 0; j < 4; ++j) {
        const float bx = smoothl1(fabsf(tg0[j] - rr0[j]))
                       + smoothl1(fabsf(tg1[j] - rr1[j]))
                       + smoothl1(fabsf(tg2[j] - rr2[j]))
                       + smoothl1(fabsf(tg3[j] - rr3[j]));
        v[j] = v[j] * mcls[j] + bx * mbox[j];
      }
      av.x = v[0] + v[1];
      av.y = v[2] + v[3];
    }
    // D[m][n] += sum_k A[m][k] * 1 : matrix pipe does the reduction tree
    acc = __builtin_amdgcn_wmma_f32_16x16x4_f32(
        false, av, false, ones, (short)0, acc, false, false);
  }

  // lane l<16 holds sum over M=0..7 of a column; lane l>=16 holds M=8..15
  const float s = acc[0] + acc[1] + acc[2] + acc[3] +
                  acc[4] + acc[5] + acc[6] + acc[7];
  const float wave_loss = __shfl(s, 0, 32) + __shfl(s, 16, 32);
#pragma unroll
  for (int off = 16; off >= 1; off >>= 1) cnt += __shfl_xor(cnt, off, 32);

  if (lane == 0) { s_loss[wave] = wave_loss; s_cnt[wave] = cnt; }
  __syncthreads();
  if (tid == 0) {
    float bl = 0.f, bc = 0.f;
#pragma unroll
    for (int w = 0; w < 8; ++w) { bl += s_loss[w]; bc += s_cnt[w]; }
    atomicAdd(&ws[b], bl);
    atomicAdd(&ws[NB + b], bc);
  }
}

__global__ void finalize_kernel(const float* __restrict__ ws,
                                float* __restrict__ out) {
  if (threadIdx.x == 0) {
    float total = 0.0f;
#pragma unroll
    for (int b = 0; b < NB; ++b) total += ws[b] / fmaxf(ws[NB + b], 1.0f);
    out[0] = total / (float)NB;
  }
}

extern "C" void kernel_launch(void* const* d_in, const int* in_sizes, int n_in,
                              void* d_out, int out_size, void* d_ws, size_t ws_size,
                              hipStream_t stream) {
  const float* dt      = (const float*)d_in[0];
  const float* gt      = (const float*)d_in[1];
  const float* anchors = (const float*)d_in[2];
  const int*   assign  = (const int*)d_in[3];
  float* ws  = (float*)d_ws;
  float* out = (float*)d_out;

  ws_init<<<1, 32, 0, stream>>>(ws);
  dim3 grid(118, NB, 1);   // 118*8 waves/batch * 128 anchors/wave >= 120000
  loss_kernel<<<grid, 256, 0, stream>>>(dt, gt, anchors, assign, ws);
  finalize_kernel<<<1, 32, 0, stream>>>(ws, out);
}